// MPNet_38293928411237
// MI455X (gfx1250) — compile-verified
//
#include <hip/hip_runtime.h>

// ---------------------------------------------------------------------------
// RGCN metapath net on gfx1250 (MI455X), wave32, V_WMMA_F32_16X16X4_F32.
//
// Per layer:
//   1) zero cnt+sum
//   2) scatter: for edges with type==rel: cnt[src]+=1; sum[src,:] += h[dst,:]
//   3) fused GEMM: out = relu( (sum/max(cnt,1)) @ W + h @ root + b )  [WMMA]
// Head: out = h3 @ lin_w + lin_b                                      [WMMA]
// ---------------------------------------------------------------------------

typedef float v2f __attribute__((ext_vector_type(2)));
typedef float v8f __attribute__((ext_vector_type(8)));

#define NODES  50000
#define DIM    96
#define LLD    64
#define NEDGES 800000

__global__ void zero_f32(float* __restrict__ p, int n) {
    int i = blockIdx.x * blockDim.x + threadIdx.x;
    if (i < n) p[i] = 0.0f;
}

// 32 lanes (one wave) per edge; each lane handles 3 of the 96 features.
__global__ void rgcn_scatter(const float* __restrict__ h,
                             const int*   __restrict__ src,
                             const int*   __restrict__ dst,
                             const int*   __restrict__ etype,
                             int rel,
                             float* __restrict__ sum,
                             float* __restrict__ cnt) {
    int gid  = blockIdx.x * blockDim.x + threadIdx.x;
    int e    = gid >> 5;
    int lane = gid & 31;
    if (e >= NEDGES) return;
    if (etype[e] != rel) return;          // only ~1/5 of edges contribute
    int s = src[e];
    int d = dst[e];
    if (lane == 0) atomicAdd(cnt + s, 1.0f);
    const float* hv = h   + (size_t)d * DIM;
    float*       sv = sum + (size_t)s * DIM;
#pragma unroll
    for (int j = lane; j < DIM; j += 32)
        atomicAdd(sv + j, hv[j]);
}

// One wave per 16x16 output tile. M=50000 (3125 tiles), N=96 (6 tiles), K=96.
// c = (sum/clip(cnt,1)) @ W + h @ root ; out = relu(c + bias)
__global__ void __launch_bounds__(256)
rgcn_gemm(const float* __restrict__ sum,
          const float* __restrict__ cnt,
          const float* __restrict__ h,
          const float* __restrict__ W,      // [96,96] row-major
          const float* __restrict__ root,   // [96,96] row-major
          const float* __restrict__ bias,   // [96]
          float* __restrict__ out) {        // [NODES,96]
    const int NT = DIM / 16;                          // 6 column tiles
    const int totalTiles = (NODES / 16) * NT;         // 18750
    int wave = (blockIdx.x * blockDim.x + threadIdx.x) >> 5;
    if (wave >= totalTiles) return;                   // wave-uniform exit

    int mt   = wave / NT;
    int nt   = wave % NT;
    int lane = threadIdx.x & 31;
    int half = lane >> 4;                             // 0 = lanes 0-15, 1 = 16-31
    int mrow = lane & 15;
    int row  = mt * 16 + mrow;                        // A-matrix row (node)
    int col  = nt * 16 + mrow;                        // B/C column

    float scale = 1.0f / fmaxf(cnt[row], 1.0f);       // mean normalization
    const float* arow = sum + (size_t)row * DIM;
    const float* hrow = h   + (size_t)row * DIM;

    v8f c = {0.f, 0.f, 0.f, 0.f, 0.f, 0.f, 0.f, 0.f};

    // agg @ W
#pragma unroll
    for (int k = 0; k < DIM; k += 4) {
        int ka = k + 2 * half;                        // ISA A/B lane layout
        v2f a, b;
        a.x = arow[ka]     * scale;
        a.y = arow[ka + 1] * scale;
        b.x = W[ka * DIM + col];
        b.y = W[(ka + 1) * DIM + col];
        c = __builtin_amdgcn_wmma_f32_16x16x4_f32(false, a, false, b,
                                                  (short)0, c, false, false);
    }
    // h @ root
#pragma unroll
    for (int k = 0; k < DIM; k += 4) {
        int ka = k + 2 * half;
        v2f a, b;
        a.x = hrow[ka];
        a.y = hrow[ka + 1];
        b.x = root[ka * DIM + col];
        b.y = root[(ka + 1) * DIM + col];
        c = __builtin_amdgcn_wmma_f32_16x16x4_f32(false, a, false, b,
                                                  (short)0, c, false, false);
    }

    float bv = bias[col];
#pragma unroll
    for (int r = 0; r < 8; ++r) {                     // C: vgpr r -> row r+8*half
        int m = mt * 16 + r + 8 * half;
        float v = c[r] + bv;
        out[(size_t)m * DIM + col] = v > 0.f ? v : 0.f;
    }
}

// Head: [50000,96] @ [96,64] + b, no ReLU. One wave per 16x16 tile.
__global__ void __launch_bounds__(256)
final_gemm(const float* __restrict__ h,
           const float* __restrict__ Wl,    // [96,64] row-major
           const float* __restrict__ bias,  // [64]
           float* __restrict__ out) {       // [NODES,64]
    const int NT = LLD / 16;                          // 4 column tiles
    const int totalTiles = (NODES / 16) * NT;         // 12500
    int wave = (blockIdx.x * blockDim.x + threadIdx.x) >> 5;
    if (wave >= totalTiles) return;

    int mt   = wave / NT;
    int nt   = wave % NT;
    int lane = threadIdx.x & 31;
    int half = lane >> 4;
    int mrow = lane & 15;
    int row  = mt * 16 + mrow;
    int col  = nt * 16 + mrow;

    const float* hrow = h + (size_t)row * DIM;

    v8f c = {0.f, 0.f, 0.f, 0.f, 0.f, 0.f, 0.f, 0.f};
#pragma unroll
    for (int k = 0; k < DIM; k += 4) {
        int ka = k + 2 * half;
        v2f a, b;
        a.x = hrow[ka];
        a.y = hrow[ka + 1];
        b.x = Wl[ka * LLD + col];
        b.y = Wl[(ka + 1) * LLD + col];
        c = __builtin_amdgcn_wmma_f32_16x16x4_f32(false, a, false, b,
                                                  (short)0, c, false, false);
    }

    float bv = bias[col];
#pragma unroll
    for (int r = 0; r < 8; ++r) {
        int m = mt * 16 + r + 8 * half;
        out[(size_t)m * LLD + col] = c[r] + bv;
    }
}

extern "C" void kernel_launch(void* const* d_in, const int* in_sizes, int n_in,
                              void* d_out, int out_size, void* d_ws, size_t ws_size,
                              hipStream_t stream) {
    const float* x     = (const float*)d_in[0];
    const int*   eidx  = (const int*)  d_in[1];   // [2, NEDGES]
    const int*   etype = (const int*)  d_in[2];   // [NEDGES]
    const float* w1    = (const float*)d_in[3];   // [5,96,96]
    const float* root1 = (const float*)d_in[4];   // [96,96]
    const float* b1    = (const float*)d_in[5];   // [96]
    const float* w2    = (const float*)d_in[6];   // [5,96,96]
    const float* root2 = (const float*)d_in[7];   // [96,96]
    const float* b2    = (const float*)d_in[8];   // [96]
    const float* lin_w = (const float*)d_in[9];   // [96,64]
    const float* lin_b = (const float*)d_in[10];  // [64]

    const int* src = eidx;                 // edge_index[0]
    const int* dst = eidx + NEDGES;        // edge_index[1]

    float* ws  = (float*)d_ws;
    float* cnt = ws;                               // NODES floats
    float* sum = ws + NODES;                       // NODES*DIM floats
    float* h1  = sum + (size_t)NODES * DIM;        // NODES*DIM floats
    float* h2  = h1  + (size_t)NODES * DIM;        // NODES*DIM floats

    const int zn = NODES + NODES * DIM;            // cnt+sum = 4,850,000 floats
    dim3 blk(256);
    dim3 zg((zn + 255) / 256);
    dim3 sg((NEDGES * 32 + 255) / 256);            // one wave per edge
    dim3 gg(((NODES / 16) * (DIM / 16) + 7) / 8);  // 8 waves per block
    dim3 fg(((NODES / 16) * (LLD / 16) + 7) / 8);

    // Layer 1: rel=0, weights w1[0], root1, b1 :  x -> h1
    zero_f32   <<<zg, blk, 0, stream>>>(cnt, zn);
    rgcn_scatter<<<sg, blk, 0, stream>>>(x, src, dst, etype, 0, sum, cnt);
    rgcn_gemm  <<<gg, blk, 0, stream>>>(sum, cnt, x, w1 + 0 * DIM * DIM,
                                        root1, b1, h1);
    // Layer 2: rel=1, weights w2[1], root2, b2 :  h1 -> h2
    zero_f32   <<<zg, blk, 0, stream>>>(cnt, zn);
    rgcn_scatter<<<sg, blk, 0, stream>>>(h1, src, dst, etype, 1, sum, cnt);
    rgcn_gemm  <<<gg, blk, 0, stream>>>(sum, cnt, h1, w2 + 1 * DIM * DIM,
                                        root2, b2, h2);
    // Layer 3: rel=2, weights w2[2], root2, b2 :  h2 -> h1 (reuse)
    zero_f32   <<<zg, blk, 0, stream>>>(cnt, zn);
    rgcn_scatter<<<sg, blk, 0, stream>>>(h2, src, dst, etype, 2, sum, cnt);
    rgcn_gemm  <<<gg, blk, 0, stream>>>(sum, cnt, h2, w2 + 2 * DIM * DIM,
                                        root2, b2, h1);
    // Head: h1 @ lin_w + lin_b -> d_out [50000,64]
    final_gemm <<<fg, blk, 0, stream>>>(h1, lin_w, lin_b, (float*)d_out);
}